// MultiHeadAttention_45672682226228
// MI455X (gfx1250) — compile-verified
//
#include <hip/hip_runtime.h>
#include <stdint.h>

typedef __attribute__((ext_vector_type(2))) float v2f;
typedef __attribute__((ext_vector_type(8))) float v8f;

#define EMBED   1024
#define HD      256      // head dim (K per head)
#define QD      64       // qkv dim (N per head)
#define OUTC    256      // 4 heads * 64
#define ROWS    16       // rows of x per workgroup
#define XPAD    4        // dword pad -> conflict-free A-fragment ds reads
#define XSTRIDE (EMBED + XPAD)   // 1028 dwords

// Async copy global -> LDS, 16 bytes per lane. GVS addressing:
// mem = SGPR64(base) + VGPR32(byte offset); LDS dest = per-lane byte address.
__device__ __forceinline__ void async_g2l_b128(uint32_t lds_addr, uint32_t goff,
                                               const void* base) {
    asm volatile("global_load_async_to_lds_b128 %0, %1, %2"
                 :: "v"(lds_addr), "v"(goff), "s"(base) : "memory");
}

__device__ __forceinline__ void async_g2l_b64(uint32_t lds_addr, uint32_t goff,
                                              const void* base) {
    asm volatile("global_load_async_to_lds_b64 %0, %1, %2"
                 :: "v"(lds_addr), "v"(goff), "s"(base) : "memory");
}

__device__ __forceinline__ void wait_async0() {
    asm volatile("s_wait_asynccnt 0x0" ::: "memory");
}

// Flat shared pointer -> 32-bit LDS byte address (ISA 10.2: LDS aperture
// maps addr[31:0] directly to the LDS address).
__device__ __forceinline__ uint32_t lds_addr_of(const void* p) {
    return (uint32_t)(uintptr_t)p;
}

// LDS:  x slab  16 x 1028 f32  = 65,792 B
//       W frags 64 k-steps x 4 q-tiles x 32 lanes x 2 f32 = 65,536 B
__global__ __launch_bounds__(256)
void mha_vproj_residual_kernel(const float* __restrict__ x,
                               const float* __restrict__ Wv,
                               float* __restrict__ out)
{
    __shared__ float xs[ROWS * XSTRIDE];
    __shared__ float wf[64 * 4 * 32 * 2];

    const int t    = threadIdx.x;     // 0..255
    const int lane = t & 31;
    const int wave = t >> 5;          // 0..7
    const long row_base = (long)blockIdx.x * ROWS;

    // ---- async coalesced fill of x slab: 256 lanes x 16B = one full row/iter
    {
        const float* src = x + row_base * EMBED;
        #pragma unroll
        for (int r = 0; r < ROWS; ++r) {
            uint32_t goff = (uint32_t)((r * EMBED + t * 4) * sizeof(float));
            uint32_t laddr = lds_addr_of(&xs[r * XSTRIDE + t * 4]);
            async_g2l_b128(laddr, goff, src);
        }
    }

    // ---- async fill of W_v pre-swizzled into B-fragment layout:
    // fragment slot flat = (k0i*4 + nb)*32 + lane ; holds Wv[q, k..k+1]
    //   q = nb*16 + (lane&15),  k = k0i*4 + 2*(lane>>4)
    {
        #pragma unroll
        for (int it = 0; it < 32; ++it) {
            int flat = it * 256 + t;        // 0..8191
            int fl   = flat & 31;
            int nb   = (flat >> 5) & 3;
            int k0i  = flat >> 7;           // 0..63
            int q = nb * 16 + (fl & 15);
            int k = k0i * 4 + 2 * (fl >> 4);
            uint32_t goff  = (uint32_t)((q * HD + k) * sizeof(float));
            uint32_t laddr = lds_addr_of(&wf[flat * 2]);
            async_g2l_b64(laddr, goff, Wv);
        }
    }

    wait_async0();          // each wave drains its own ASYNCcnt
    __syncthreads();        // then all waves agree LDS is filled

    // ---- each wave: two adjacent 16x16 tiles (same head -> shared A fragment)
    const int ct0   = wave * 2;        // column tile 0..15 (16 cols each)
    const int h     = ct0 >> 2;        // head 0..3
    const int kbase = h * HD;          // x column base for this head
    const int nb0   = ct0 & 3;         // q-tile within head for tile 0
    const int nb1   = (ct0 + 1) & 3;   // ... tile 1 (same head, next q-tile)

    v8f acc0 = {0.f, 0.f, 0.f, 0.f, 0.f, 0.f, 0.f, 0.f};
    v8f acc1 = {0.f, 0.f, 0.f, 0.f, 0.f, 0.f, 0.f, 0.f};

    const int am  = lane & 15;         // A-fragment row (M)
    const int ak2 = 2 * (lane >> 4);   // A-fragment K offset (0 or 2)

    #pragma unroll 8
    for (int k0i = 0; k0i < 64; ++k0i) {
        int k0 = k0i * 4;
        v2f a  = *(const v2f*)(&xs[am * XSTRIDE + kbase + k0 + ak2]);
        v2f b0 = *(const v2f*)(&wf[((k0i * 4 + nb0) * 32 + lane) * 2]);
        v2f b1 = *(const v2f*)(&wf[((k0i * 4 + nb1) * 32 + lane) * 2]);
        acc0 = __builtin_amdgcn_wmma_f32_16x16x4_f32(
                   false, a, false, b0, (short)0, acc0, false, false);
        acc1 = __builtin_amdgcn_wmma_f32_16x16x4_f32(
                   false, a, false, b1, (short)0, acc1, false, false);
    }

    // ---- epilogue: out[m, c] = acc + x[m, 768 + c]  (residual from LDS slab)
    // C/D layout: VGPR j, lane L -> M = j + 8*(L>>4), N = L&15
    const int mhi  = (lane >> 4) * 8;
    const int ncol = lane & 15;
    const int c0   = ct0 * 16 + ncol;
    const int c1   = c0 + 16;
    float* outp = out + row_base * OUTC;

    #pragma unroll
    for (int j = 0; j < 8; ++j) {
        int m = mhi + j;
        float r0 = xs[m * XSTRIDE + 3 * HD + c0];
        float r1 = xs[m * XSTRIDE + 3 * HD + c1];
        outp[(long)m * OUTC + c0] = acc0[j] + r0;
        outp[(long)m * OUTC + c1] = acc1[j] + r1;
    }
}

extern "C" void kernel_launch(void* const* d_in, const int* in_sizes, int n_in,
                              void* d_out, int out_size, void* d_ws, size_t ws_size,
                              hipStream_t stream) {
    // setup_inputs order: x, W_q, W_k, W_v  (Q/K feed only dead code)
    const float* x  = (const float*)d_in[0];
    const float* Wv = (const float*)d_in[3];
    float* out = (float*)d_out;

    const int total_rows = 16 * 4096;         // B * S
    dim3 grid(total_rows / ROWS), block(256);
    hipLaunchKernelGGL(mha_vproj_residual_kernel, grid, block, 0, stream,
                       x, Wv, out);
}